// RandomWalkNet_790273982776
// MI455X (gfx1250) — compile-verified
//
#include <hip/hip_runtime.h>
#include <hip/hip_bf16.h>

// ---- sizes (compile-time; instances_num == 4 per reference setup) ----
#define N_ROWS 4096
#define D_IN   2048
#define C_DIM  256
#define Q      1024   // probes
#define G      3072   // gallery
#define ALPHA  0.5f
#define ITERS  40     // 0.5^40 ~ 1e-12 -> fp32 round-off

// k_iter LDS tiling
#define KC     32            // K chunk
#define LDS_STRIDE 36        // 32 + 4 pad: 16B-aligned rows, conflict-free banks
#define TILE_FLOATS (128 * LDS_STRIDE)
#define SMEM_BYTES (4 * TILE_FLOATS * 4)   // lA[2] + lB[2]

typedef float v2f __attribute__((ext_vector_type(2)));
typedef float v8f __attribute__((ext_vector_type(8)));

static __device__ __forceinline__ v8f wmma4(v2f a, v2f b, v8f c) {
    // V_WMMA_F32_16X16X4_F32 : D = A(16x4) x B(4x16) + C(16x16), all f32
    return __builtin_amdgcn_wmma_f32_16x16x4_f32(
        /*neg_a=*/false, a, /*neg_b=*/false, b,
        /*c_mod=*/(short)0, c, /*reuse_a=*/false, /*reuse_b=*/false);
}

// ---- CDNA5 async copy: global -> LDS, 16B per lane, ASYNCcnt tracked ----
static __device__ __forceinline__ void async_b128(unsigned lds_off, const float* gsrc) {
    asm volatile("global_load_async_to_lds_b128 %0, %1, off"
                 :: "v"(lds_off), "v"(gsrc) : "memory");
}
static __device__ __forceinline__ void wait_async0() {
    asm volatile("s_wait_asynccnt 0x0" ::: "memory");
}

// copy a 128-row x KC-float tile (row stride G in global, LDS_STRIDE in LDS)
static __device__ __forceinline__ void copy_tile(unsigned lds_base, const float* gbase, int tid) {
#pragma unroll
    for (int i = 0; i < (128 * KC / 4) / 256; ++i) {   // 4 b128 per thread
        int idx4 = tid + i * 256;                      // 0..1023
        int row  = idx4 >> 3;                          // KC/4 = 8 float4 per row
        int col  = (idx4 & 7) << 2;                    // float col
        unsigned dst = lds_base + (unsigned)(row * LDS_STRIDE + col) * 4u;
        async_b128(dst, gbase + (size_t)row * G + col);
    }
}

// ------------------------------------------------------------------
// K1: feat = x @ W_base + b_base     (4096 x 2048 x 256)
// block tile 64x64, 8 waves (4M x 2N), wave tile 16x32
// ------------------------------------------------------------------
__global__ __launch_bounds__(256) void k_feat(const float* __restrict__ X,
                                              const float* __restrict__ W,
                                              const float* __restrict__ bias,
                                              float* __restrict__ feat) {
    int tid = threadIdx.x, lane = tid & 31, wave = tid >> 5;
    int l16 = lane & 15, half = lane >> 4;
    int m0 = blockIdx.x * 64 + (wave >> 1) * 16;
    int nb = blockIdx.y * 64 + (wave & 1) * 32;
    v8f c0 = {}, c1 = {};
    const float* arow = X + (size_t)(m0 + l16) * D_IN;
    for (int k = 0; k < D_IN; k += 4) {
        int k0 = k + 2 * half;
        v2f a = { arow[k0], arow[k0 + 1] };
        v2f b0 = { W[(size_t)k0 * C_DIM + nb + l16],
                   W[(size_t)(k0 + 1) * C_DIM + nb + l16] };
        v2f b1 = { W[(size_t)k0 * C_DIM + nb + 16 + l16],
                   W[(size_t)(k0 + 1) * C_DIM + nb + 16 + l16] };
        c0 = wmma4(a, b0, c0);
        c1 = wmma4(a, b1, c1);
    }
#pragma unroll
    for (int v = 0; v < 8; ++v) {
        int m = m0 + v + 8 * half;
        int n = nb + l16;
        feat[(size_t)m * C_DIM + n]      = c0[v] + bias[n];
        feat[(size_t)m * C_DIM + n + 16] = c1[v] + bias[n + 16];
    }
}

// ------------------------------------------------------------------
// K2: gather probe/gallery rows, GWk = gallery .* W_embed[:,k],
//     s[i,k] = sum_d feat[i,d]^2 * W_embed[d,k]
// ------------------------------------------------------------------
__global__ __launch_bounds__(256) void k_gather(const float* __restrict__ feat,
                                                const float* __restrict__ W_embed,
                                                float* __restrict__ probe,
                                                float* __restrict__ gallery,
                                                float* __restrict__ GW0,
                                                float* __restrict__ GW1,
                                                float* __restrict__ s_p,
                                                float* __restrict__ s_g) {
    int b = blockIdx.x;           // feat row
    int d = threadIdx.x;          // 0..255
    int grp = b >> 2, slot = b & 3;
    float g  = feat[(size_t)b * C_DIM + d];
    float w0 = W_embed[d * 2 + 0];
    float w1 = W_embed[d * 2 + 1];
    __shared__ float red0[256], red1[256];
    red0[d] = g * g * w0;
    red1[d] = g * g * w1;
    if (slot == 0) {
        probe[(size_t)grp * C_DIM + d] = g;
    } else {
        int j = grp * 3 + (slot - 1);
        gallery[(size_t)j * C_DIM + d] = g;
        GW0[(size_t)j * C_DIM + d] = g * w0;
        GW1[(size_t)j * C_DIM + d] = g * w1;
    }
    __syncthreads();
    for (int s = 128; s >= 1; s >>= 1) {
        if (d < s) { red0[d] += red0[d + s]; red1[d] += red1[d + s]; }
        __syncthreads();
    }
    if (d == 0) {
        if (slot == 0) { s_p[grp * 2 + 0] = red0[0]; s_p[grp * 2 + 1] = red1[0]; }
        else { int j = grp * 3 + slot - 1; s_g[j * 2 + 0] = red0[0]; s_g[j * 2 + 1] = red1[0]; }
    }
}

// ------------------------------------------------------------------
// K3: g_g assemble (M=N=3072, K=256), both channels in one pass.
// ------------------------------------------------------------------
__global__ __launch_bounds__(256) void k_gg(const float* __restrict__ gallery,
                                            const float* __restrict__ GW0,
                                            const float* __restrict__ GW1,
                                            const float* __restrict__ s_g,
                                            const float* __restrict__ b_embed,
                                            float* __restrict__ out_g,
                                            float* __restrict__ G1) {
    int tid = threadIdx.x, lane = tid & 31, wave = tid >> 5;
    int l16 = lane & 15, half = lane >> 4;
    int m0 = blockIdx.x * 64 + (wave >> 1) * 16;
    int nb = blockIdx.y * 64 + (wave & 1) * 32;
    v8f c00 = {}, c01 = {}, c10 = {}, c11 = {};
    const float* arow = gallery + (size_t)(m0 + l16) * C_DIM;
    for (int k = 0; k < C_DIM; k += 4) {
        int k0 = k + 2 * half;
        v2f a = { arow[k0], arow[k0 + 1] };
        const float* p00 = GW0 + (size_t)(nb + l16) * C_DIM + k0;
        const float* p01 = GW0 + (size_t)(nb + 16 + l16) * C_DIM + k0;
        const float* p10 = GW1 + (size_t)(nb + l16) * C_DIM + k0;
        const float* p11 = GW1 + (size_t)(nb + 16 + l16) * C_DIM + k0;
        v2f b;
        b = (v2f){ p00[0], p00[1] }; c00 = wmma4(a, b, c00);
        b = (v2f){ p01[0], p01[1] }; c01 = wmma4(a, b, c01);
        b = (v2f){ p10[0], p10[1] }; c10 = wmma4(a, b, c10);
        b = (v2f){ p11[0], p11[1] }; c11 = wmma4(a, b, c11);
    }
    float be0 = b_embed[0], be1 = b_embed[1];
#pragma unroll
    for (int v = 0; v < 8; ++v) {
        int i = m0 + v + 8 * half;
        float si0 = s_g[i * 2 + 0], si1 = s_g[i * 2 + 1];
        {   int j = nb + l16;
            float e0 = si0 + s_g[j * 2 + 0] - 2.0f * c00[v] + be0;
            float e1 = si1 + s_g[j * 2 + 1] - 2.0f * c10[v] + be1;
            *(float2*)(out_g + ((size_t)i * G + j) * 2) = make_float2(e0, e1);
            G1[(size_t)i * G + j] = e1;
        }
        {   int j = nb + 16 + l16;
            float e0 = si0 + s_g[j * 2 + 0] - 2.0f * c01[v] + be0;
            float e1 = si1 + s_g[j * 2 + 1] - 2.0f * c11[v] + be1;
            *(float2*)(out_g + ((size_t)i * G + j) * 2) = make_float2(e0, e1);
            G1[(size_t)i * G + j] = e1;
        }
    }
}

// ------------------------------------------------------------------
// K4: p_g assemble (M=1024, N=3072, K=256); channel-0 to out,
//     P1 packed, Y0 = (1-alpha)*P1.
// ------------------------------------------------------------------
__global__ __launch_bounds__(256) void k_pg(const float* __restrict__ probe,
                                            const float* __restrict__ GW0,
                                            const float* __restrict__ GW1,
                                            const float* __restrict__ s_p,
                                            const float* __restrict__ s_g,
                                            const float* __restrict__ b_embed,
                                            float* __restrict__ out_p,
                                            float* __restrict__ P1,
                                            float* __restrict__ Y0) {
    int tid = threadIdx.x, lane = tid & 31, wave = tid >> 5;
    int l16 = lane & 15, half = lane >> 4;
    int m0 = blockIdx.x * 64 + (wave >> 1) * 16;
    int nb = blockIdx.y * 64 + (wave & 1) * 32;
    v8f c00 = {}, c01 = {}, c10 = {}, c11 = {};
    const float* arow = probe + (size_t)(m0 + l16) * C_DIM;
    for (int k = 0; k < C_DIM; k += 4) {
        int k0 = k + 2 * half;
        v2f a = { arow[k0], arow[k0 + 1] };
        const float* p00 = GW0 + (size_t)(nb + l16) * C_DIM + k0;
        const float* p01 = GW0 + (size_t)(nb + 16 + l16) * C_DIM + k0;
        const float* p10 = GW1 + (size_t)(nb + l16) * C_DIM + k0;
        const float* p11 = GW1 + (size_t)(nb + 16 + l16) * C_DIM + k0;
        v2f b;
        b = (v2f){ p00[0], p00[1] }; c00 = wmma4(a, b, c00);
        b = (v2f){ p01[0], p01[1] }; c01 = wmma4(a, b, c01);
        b = (v2f){ p10[0], p10[1] }; c10 = wmma4(a, b, c10);
        b = (v2f){ p11[0], p11[1] }; c11 = wmma4(a, b, c11);
    }
    float be0 = b_embed[0], be1 = b_embed[1];
#pragma unroll
    for (int v = 0; v < 8; ++v) {
        int p = m0 + v + 8 * half;
        float si0 = s_p[p * 2 + 0], si1 = s_p[p * 2 + 1];
        {   int j = nb + l16;
            float e0 = si0 + s_g[j * 2 + 0] - 2.0f * c00[v] + be0;
            float e1 = si1 + s_g[j * 2 + 1] - 2.0f * c10[v] + be1;
            out_p[((size_t)p * G + j) * 2] = e0;
            P1[(size_t)p * G + j] = e1;
            Y0[(size_t)p * G + j] = (1.0f - ALPHA) * e1;
        }
        {   int j = nb + 16 + l16;
            float e0 = si0 + s_g[j * 2 + 0] - 2.0f * c01[v] + be0;
            float e1 = si1 + s_g[j * 2 + 1] - 2.0f * c11[v] + be1;
            out_p[((size_t)p * G + j) * 2] = e0;
            P1[(size_t)p * G + j] = e1;
            Y0[(size_t)p * G + j] = (1.0f - ALPHA) * e1;
        }
    }
}

// ------------------------------------------------------------------
// K5: rsinv[i] = 1 / (sum_j G1[i,j] - G1[i,i])
// ------------------------------------------------------------------
__global__ __launch_bounds__(256) void k_rowsum(const float* __restrict__ G1,
                                                float* __restrict__ rsinv) {
    int i = blockIdx.x, t = threadIdx.x;
    const float* row = G1 + (size_t)i * G;
    float s = 0.0f;
    for (int j = t; j < G; j += 256) s += row[j];
    __shared__ float red[256];
    red[t] = s; __syncthreads();
    for (int st = 128; st >= 1; st >>= 1) {
        if (t < st) red[t] += red[t + st];
        __syncthreads();
    }
    if (t == 0) rsinv[i] = 1.0f / (red[0] - row[i]);
}

// ------------------------------------------------------------------
// K6: Richardson step (hot loop, 40x; G1 + Y are L2-resident):
//   Ynew[p,i] = (1-a)*P1[p,i] + a*rsinv[i] * sum_j Yold[p,j]*G1[i,j]
// GEMM M=1024, N=3072, K=3072. Block tile 128x128 (8 waves 4Mx2N,
// wave tile 32x64). Double-buffered LDS tiles filled with
// GLOBAL_LOAD_ASYNC_TO_LDS_B128 (ASYNCcnt), fragments via ds_load_b64.
// ------------------------------------------------------------------
__global__ __launch_bounds__(256) void k_iter(const float* __restrict__ Yold,
                                              const float* __restrict__ G1m,
                                              const float* __restrict__ P1,
                                              const float* __restrict__ rsinv,
                                              float* __restrict__ Ynew) {
    extern __shared__ float smem[];    // lA[2] | lB[2], TILE_FLOATS each
    int tid = threadIdx.x, lane = tid & 31, wave = tid >> 5;
    int l16 = lane & 15, half = lane >> 4;
    int mw = wave >> 1, nw = wave & 1;
    int mBase = blockIdx.x * 128, nBase = blockIdx.y * 128;

    unsigned ldsA = (unsigned)(uintptr_t)&smem[0];
    unsigned ldsB = (unsigned)(uintptr_t)&smem[2 * TILE_FLOATS];
    const float* gA = Yold + (size_t)mBase * G;
    const float* gB = G1m  + (size_t)nBase * G;

    v8f acc[2][4] = {};

    copy_tile(ldsA, gA, tid);          // chunk 0 -> buffer 0
    copy_tile(ldsB, gB, tid);
    wait_async0();
    __syncthreads();

    for (int kc = 0; kc < G / KC; ++kc) {
        int cur = kc & 1, nxt = cur ^ 1;
        if (kc + 1 < G / KC) {         // prefetch chunk kc+1 into other buffer
            copy_tile(ldsA + (unsigned)nxt * TILE_FLOATS * 4u, gA + (kc + 1) * KC, tid);
            copy_tile(ldsB + (unsigned)nxt * TILE_FLOATS * 4u, gB + (kc + 1) * KC, tid);
        }
        const float* As = smem + (size_t)cur * TILE_FLOATS;
        const float* Bs = smem + (size_t)(2 + cur) * TILE_FLOATS;
#pragma unroll
        for (int k = 0; k < KC; k += 4) {
            int k0 = k + 2 * half;
            const float* a0p = &As[(mw * 32 + l16) * LDS_STRIDE + k0];
            const float* a1p = &As[(mw * 32 + 16 + l16) * LDS_STRIDE + k0];
            v2f a0 = { a0p[0], a0p[1] };                   // ds_load_b64
            v2f a1 = { a1p[0], a1p[1] };
#pragma unroll
            for (int ns = 0; ns < 4; ++ns) {
                const float* bp = &Bs[(nw * 64 + ns * 16 + l16) * LDS_STRIDE + k0];
                v2f b = { bp[0], bp[1] };
                acc[0][ns] = wmma4(a0, b, acc[0][ns]);
                acc[1][ns] = wmma4(a1, b, acc[1][ns]);
            }
        }
        wait_async0();
        __syncthreads();
    }

#pragma unroll
    for (int ms = 0; ms < 2; ++ms)
#pragma unroll
    for (int ns = 0; ns < 4; ++ns) {
        int i = nBase + nw * 64 + ns * 16 + l16;
        float ri = ALPHA * rsinv[i];
#pragma unroll
        for (int v = 0; v < 8; ++v) {
            int p = mBase + mw * 32 + ms * 16 + v + 8 * half;
            size_t o = (size_t)p * G + i;
            Ynew[o] = (1.0f - ALPHA) * P1[o] + ri * acc[ms][ns][v];
        }
    }
}

// ------------------------------------------------------------------
// K7: scatter solved channel-1 into interleaved p_g output
// ------------------------------------------------------------------
__global__ __launch_bounds__(256) void k_final(const float* __restrict__ Y,
                                               float* __restrict__ out_p) {
    size_t idx = (size_t)blockIdx.x * 256 + threadIdx.x;   // Q*G total
    out_p[idx * 2 + 1] = Y[idx];
}

// ------------------------------------------------------------------
extern "C" void kernel_launch(void* const* d_in, const int* in_sizes, int n_in,
                              void* d_out, int out_size, void* d_ws, size_t ws_size,
                              hipStream_t stream) {
    const float* x       = (const float*)d_in[0];
    const float* W_base  = (const float*)d_in[1];
    const float* b_base  = (const float*)d_in[2];
    const float* W_embed = (const float*)d_in[3];
    const float* b_embed = (const float*)d_in[4];
    float* out = (float*)d_out;
    float* ws  = (float*)d_ws;

    float* feat    = ws;                                 // 4096*256
    float* probe   = feat    + (size_t)N_ROWS * C_DIM;   // 1024*256
    float* gallery = probe   + (size_t)Q * C_DIM;        // 3072*256
    float* GW0     = gallery + (size_t)G * C_DIM;
    float* GW1     = GW0     + (size_t)G * C_DIM;
    float* s_p     = GW1     + (size_t)G * C_DIM;        // 1024*2
    float* s_g     = s_p     + (size_t)Q * 2;            // 3072*2
    float* rsinv   = s_g     + (size_t)G * 2;            // 3072
    float* G1      = rsinv   + (size_t)G;                // 3072*3072
    float* P1      = G1      + (size_t)G * G;            // 1024*3072
    float* Y0      = P1      + (size_t)Q * G;
    float* Y1      = Y0      + (size_t)Q * G;

    float* out_p = out;                                  // (1024*3072, 2)
    float* out_g = out + (size_t)Q * G * 2;              // (3072*3072, 2)

    k_feat  <<<dim3(N_ROWS / 64, C_DIM / 64), 256, 0, stream>>>(x, W_base, b_base, feat);
    k_gather<<<N_ROWS, 256, 0, stream>>>(feat, W_embed, probe, gallery, GW0, GW1, s_p, s_g);
    k_gg    <<<dim3(G / 64, G / 64), 256, 0, stream>>>(gallery, GW0, GW1, s_g, b_embed, out_g, G1);
    k_pg    <<<dim3(Q / 64, G / 64), 256, 0, stream>>>(probe, GW0, GW1, s_p, s_g, b_embed, out_p, P1, Y0);
    k_rowsum<<<G, 256, 0, stream>>>(G1, rsinv);

    const float* Yin = Y0;
    float* Yout = Y1;
    for (int t = 0; t < ITERS; ++t) {
        k_iter<<<dim3(Q / 128, G / 128), 256, SMEM_BYTES, stream>>>(Yin, G1, P1, rsinv, Yout);
        const float* tmp = Yout; Yout = (float*)Yin; Yin = tmp;
    }
    k_final<<<(Q * G) / 256, 256, 0, stream>>>(Yin, out_p);
}